// WrapperYukarinSa_39530878992779
// MI455X (gfx1250) — compile-verified
//
#include <hip/hip_runtime.h>

// ---------------------------------------------------------------------------
// YukarinSa pitch model for MI455X (gfx1250, wave32, WGP, WMMA, TDM)
//   K1: gather/concat -> enc_in
//   K2: fp32 WMMA GEMM (16x16x4), 32x64 tile/wave, 8 accumulators
//   K3: persistent-gang GRU scan; Whh staged to LDS via Tensor Data Mover
//       (hardware pad: 4 dwords / 256 dwords -> conflict-free 1040 stride)
// ---------------------------------------------------------------------------

typedef __attribute__((ext_vector_type(2))) float    v2f;
typedef __attribute__((ext_vector_type(8))) float    v8f;
typedef __attribute__((ext_vector_type(4))) unsigned ui32x4;
typedef __attribute__((ext_vector_type(8))) int      i32x8;
typedef __attribute__((ext_vector_type(4))) int      i32x4;

#define L_SEQ   8192
#define E_DIM   256
#define CIN     516          // E + 4 + E
#define HDIM    1024
#define G3      3072         // 3*HDIM
#define LDW_IH  1025         // gru_Wih row stride (H+1)

#define NWG           64
#define UNITS_PER_WG  16     // 1024 / 64
#define TPW           256    // threads per scan workgroup
#define USTRIDE       1040   // LDS row stride: 1024 + 4dw pad per 256dw (TDM pad)

// ---- fp32 WMMA 16x16x4 ----------------------------------------------------
__device__ __forceinline__ v8f wmma_f32_16x16x4(v2f a, v2f b, v8f c) {
#if __has_builtin(__builtin_amdgcn_wmma_f32_16x16x4_f32)
  return __builtin_amdgcn_wmma_f32_16x16x4_f32(false, a, false, b,
                                               (short)0, c, false, false);
#else
  c[0] += a.x * b.x + a.y * b.y;   // compile-only safety net
  return c;
#endif
}

// LDS layout used by both TDM hardware padding and the compute loop:
// element k of unit-row u lives at  u*USTRIDE + k + (k>>8)*4
__device__ __forceinline__ int lds_off(int k) { return k + ((k >> 8) << 2); }

// ---- K1: build enc_in = [ph_emb(v+1)+ph_emb(c+1) | accents(4) | spk(256)] --
__global__ void build_enc_in(const int* __restrict__ vow, const int* __restrict__ con,
                             const int* __restrict__ sa,  const int* __restrict__ ea,
                             const int* __restrict__ sap, const int* __restrict__ eap,
                             const int* __restrict__ spk_id,
                             const float* __restrict__ ph_emb,
                             const float* __restrict__ spk_emb,
                             float* __restrict__ enc_in) {
  const int l = blockIdx.x;
  const int j = threadIdx.x;                         // 0..255
  const int v = vow[l] + 1, c = con[l] + 1;
  const float ph = ph_emb[(size_t)v * E_DIM + j] + ph_emb[(size_t)c * E_DIM + j];
  const float* spk = spk_emb + (size_t)spk_id[0] * E_DIM;
  float* row = enc_in + (size_t)l * CIN;
  row[j] = ph;
  row[E_DIM + 4 + j] = spk[j];
  if (j < 4) {
    int a = (j == 0) ? sa[l] : (j == 1) ? ea[l] : (j == 2) ? sap[l] : eap[l];
    row[E_DIM + j] = (float)a;
  }
}

// ---- K2: C[M,N] = A[M,K] @ W[N,K]^T (+bias[n]) -----------------------------
// 256 threads = 8 waves arranged 2(M) x 4(N); each wave owns a 32x64 tile
// -> per K-step: 2 A-frags + 4 B-frags feed 8 v_wmma_f32_16x16x4_f32.
__global__ void wmma_gemm_nt(const float* __restrict__ A, int lda,
                             const float* __restrict__ W, int ldw,
                             const float* __restrict__ bias,
                             float* __restrict__ C, int ldc, int K) {
  const int lane = threadIdx.x & 31;
  const int wave = threadIdx.x >> 5;
  const int wm = wave >> 2;                          // 0..1
  const int wn = wave & 3;                           // 0..3
  const int m0 = blockIdx.x * 64 + wm * 32;
  const int n0 = blockIdx.y * 256 + wn * 64;
  const int r  = lane & 15;
  const int kh = (lane >> 4) * 2;                    // K pair {0,1} or {2,3}

  const float* Ap[2] = { A + (size_t)(m0 + r) * lda,
                         A + (size_t)(m0 + 16 + r) * lda };
  const float* Wp[4] = { W + (size_t)(n0 +  0 + r) * ldw,
                         W + (size_t)(n0 + 16 + r) * ldw,
                         W + (size_t)(n0 + 32 + r) * ldw,
                         W + (size_t)(n0 + 48 + r) * ldw };
  v8f acc[2][4] = {};
  for (int k = 0; k < K; k += 4) {
    v2f a[2], b[4];
#pragma unroll
    for (int tm = 0; tm < 2; ++tm) { a[tm].x = Ap[tm][k + kh]; a[tm].y = Ap[tm][k + kh + 1]; }
#pragma unroll
    for (int tn = 0; tn < 4; ++tn) { b[tn].x = Wp[tn][k + kh]; b[tn].y = Wp[tn][k + kh + 1]; }
#pragma unroll
    for (int tm = 0; tm < 2; ++tm)
#pragma unroll
      for (int tn = 0; tn < 4; ++tn)
        acc[tm][tn] = wmma_f32_16x16x4(a[tm], b[tn], acc[tm][tn]);
  }
#pragma unroll
  for (int tm = 0; tm < 2; ++tm) {
    const int mb = m0 + tm * 16 + ((lane >> 4) << 3);
#pragma unroll
    for (int tn = 0; tn < 4; ++tn) {
      const int n = n0 + tn * 16 + r;
      const float bv = bias ? bias[n] : 0.0f;
#pragma unroll
      for (int e = 0; e < 8; ++e)
        C[(size_t)(mb + e) * ldc + n] = acc[tm][tn][e] + bv;
    }
  }
}

// ---- init: zero state, barrier counter, f0 ---------------------------------
__global__ void init_state(float* g_hbuf, float* g_f0, unsigned* bar) {
  const int t = threadIdx.x;
  g_hbuf[t] = 0.0f;
  g_hbuf[HDIM + t] = 0.0f;
  if (t == 0) { *g_f0 = 0.0f; *bar = 0u; }
}

// ---- device-wide gang barrier (monotonic counter; replay-safe via init) ----
__device__ __forceinline__ void gang_barrier(unsigned* bar) {
  __syncthreads();
  if (threadIdx.x == 0) {
    __threadfence();                                   // release
    unsigned old = atomicAdd(bar, 1u);
    unsigned target = (old / NWG + 1u) * NWG;
    while (atomicAdd(bar, 0u) < target)
      __builtin_amdgcn_s_sleep(1);
    __threadfence();                                   // acquire
  }
  __syncthreads();
}

// ---- K3: sequential GRU scan; Whh slice LDS-resident per WGP ---------------
__global__ void gru_scan(const float* __restrict__ gi_h,   // [L][3072]
                         const float* __restrict__ Whh,    // [3072][1024]
                         const float* __restrict__ Wih,    // [3072][1025] (f0 column)
                         const float* __restrict__ bih,    // [3072]
                         const float* __restrict__ bhh,    // [3072]
                         const float* __restrict__ postW,  // [1024]
                         const float* __restrict__ postb,  // [1]
                         float* __restrict__ g_hbuf,       // [2][1024] ping-pong
                         float* __restrict__ g_f0,
                         unsigned* __restrict__ bar,
                         float* __restrict__ out) {
  extern __shared__ float smem[];
  float* sW = smem;                                     // 3 * 16 * USTRIDE
  float* sh = smem + 3 * UNITS_PER_WG * USTRIDE;        // 1024
  __shared__ float red[8];

  const int tid = threadIdx.x;
  const int wg  = blockIdx.x;
  const int u   = tid >> 4;                             // local unit 0..15
  const int p   = tid & 15;                             // K-partition 0..15
  const int i   = wg * UNITS_PER_WG + u;                // global hidden unit

  // ---- one-time stage of this WGP's Whh slice into LDS ---------------------
#if __has_builtin(__builtin_amdgcn_tensor_load_to_lds)
  if (tid < 32) {                                       // wave 0 drives the TDM
    const unsigned ldsBase = (unsigned)(uintptr_t)sW;   // LDS byte offset (low 32)
#pragma unroll
    for (int g = 0; g < 3; ++g) {
      const unsigned long long ga = (unsigned long long)(uintptr_t)
          (Whh + ((size_t)g * HDIM + (size_t)wg * UNITS_PER_WG) * HDIM);
      ui32x4 g0;
      g0[0] = 1u;                                       // count=1 user descriptor
      g0[1] = ldsBase + (unsigned)(g * UNITS_PER_WG * USTRIDE * 4);
      g0[2] = (unsigned)(ga & 0xffffffffu);             // global_addr[31:0]
      g0[3] = (unsigned)((ga >> 32) & 0x01ffffffu)      // global_addr[56:32]
              | (2u << 30);                             // type = 2 ("image")
      i32x8 g1;
      g1[0] = (int)((2u << 16)                          // data_size = 4B
                    | (1u << 20)                        // pad_enable
                    | (7u << 22)                        // pad_interval = 256 dw
                    | (3u << 25));                      // pad_amount   = 4 dw
      g1[1] = (int)(HDIM << 16);                        // tensor_dim0 = 1024
      g1[2] = (int)(UNITS_PER_WG << 16);                // tensor_dim1 = 16
      g1[3] = (int)(HDIM << 16);                        // tile_dim0   = 1024
      g1[4] = (int)UNITS_PER_WG;                        // tile_dim1=16, tile_dim2=0
      g1[5] = (int)HDIM;                                // tensor_dim0_stride = 1024
      g1[6] = 0; g1[7] = 0;
      const i32x4 gz4 = {0, 0, 0, 0};
      const i32x8 gz8 = {0, 0, 0, 0, 0, 0, 0, 0};
      // 6-arg form (clang-23 / therock-10.0 headers): groups 0..3 + extra + cpol
      __builtin_amdgcn_tensor_load_to_lds(g0, g1, gz4, gz4, gz8, 0);
    }
    __builtin_amdgcn_s_wait_tensorcnt(0);
  }
#else
  for (int idx = tid; idx < UNITS_PER_WG * HDIM; idx += TPW) {
    const int uu = idx >> 10, k = idx & (HDIM - 1);
    const int gr = wg * UNITS_PER_WG + uu;
#pragma unroll
    for (int g = 0; g < 3; ++g)
      sW[(g * UNITS_PER_WG + uu) * USTRIDE + lds_off(k)] =
          Whh[(size_t)(g * HDIM + gr) * HDIM + k];
  }
#endif
  const float wf0_r = Wih[(size_t)(i)            * LDW_IH + HDIM];
  const float wf0_z = Wih[(size_t)(HDIM + i)     * LDW_IH + HDIM];
  const float wf0_n = Wih[(size_t)(2 * HDIM + i) * LDW_IH + HDIM];
  const float bi_r = bih[i], bi_z = bih[HDIM + i], bi_n = bih[2 * HDIM + i];
  const float bh_r = bhh[i], bh_z = bhh[HDIM + i], bh_n = bhh[2 * HDIM + i];
  const float* wr = sW + (0 * UNITS_PER_WG + u) * USTRIDE;
  const float* wz = sW + (1 * UNITS_PER_WG + u) * USTRIDE;
  const float* wn = sW + (2 * UNITS_PER_WG + u) * USTRIDE;
  __syncthreads();

  for (int t = 0; t < L_SEQ; ++t) {
    const float* hR = g_hbuf + ((t & 1) ? HDIM : 0);    // read buffer
    float*       hW = g_hbuf + ((t & 1) ? 0 : HDIM);    // write buffer
    const float f0 = *g_f0;
    if (p == 0 && t + 1 < L_SEQ)                        // hide next step's gi row
      __builtin_prefetch(gi_h + (size_t)(t + 1) * G3 + i, 0, 1);
    for (int idx = tid; idx < HDIM; idx += TPW) sh[idx] = hR[idx];
    __syncthreads();

    float ar = 0.f, az = 0.f, an = 0.f;
#pragma unroll 8
    for (int j = 0; j < HDIM / 16; ++j) {               // k = j*16 + p : bank-spread
      const int k = j * 16 + p;
      const int o = lds_off(k);
      const float hk = sh[k];
      ar = fmaf(wr[o], hk, ar);
      az = fmaf(wz[o], hk, az);
      an = fmaf(wn[o], hk, an);
    }
    for (int o = 8; o; o >>= 1) {                       // reduce 16 K-partitions
      ar += __shfl_xor(ar, o, 32);
      az += __shfl_xor(az, o, 32);
      an += __shfl_xor(an, o, 32);
    }
    if (p == 0) {
      const float* gi = gi_h + (size_t)t * G3;
      const float gr  = gi[i]            + f0 * wf0_r + bi_r + ar + bh_r;
      const float gz  = gi[HDIM + i]     + f0 * wf0_z + bi_z + az + bh_z;
      const float ghn = an + bh_n;
      const float gn  = gi[2 * HDIM + i] + f0 * wf0_n + bi_n;
      const float rg = 1.0f / (1.0f + __expf(-gr));
      const float zg = 1.0f / (1.0f + __expf(-gz));
      const float ng = tanhf(gn + rg * ghn);
      hW[i] = (1.0f - zg) * ng + zg * sh[i];
    }
    gang_barrier(bar);                                  // all h_new visible

    if (wg == 0) {                                      // f0 = postW . h_new + b
      float part = 0.f;
      for (int idx = tid; idx < HDIM; idx += TPW)
        part = fmaf(postW[idx], hW[idx], part);
      for (int o = 16; o; o >>= 1) part += __shfl_xor(part, o, 32);
      if ((tid & 31) == 0) red[tid >> 5] = part;
      __syncthreads();
      if (tid == 0) {
        float s = 0.f;
        for (int w = 0; w < 8; ++w) s += red[w];
        const float f0n = s + postb[0];
        *g_f0 = f0n;
        out[t] = f0n;
      }
    }
    gang_barrier(bar);                                  // f0 visible before next step
    __syncthreads();
  }
}

// ---------------------------------------------------------------------------
extern "C" void kernel_launch(void* const* d_in, const int* in_sizes, int n_in,
                              void* d_out, int out_size, void* d_ws, size_t ws_size,
                              hipStream_t stream) {
  (void)in_sizes; (void)n_in; (void)out_size; (void)ws_size;
  const int*   vow    = (const int*)d_in[1];
  const int*   con    = (const int*)d_in[2];
  const int*   sa     = (const int*)d_in[3];
  const int*   ea     = (const int*)d_in[4];
  const int*   sap    = (const int*)d_in[5];
  const int*   eap    = (const int*)d_in[6];
  const int*   spkid  = (const int*)d_in[7];
  const float* ph_emb = (const float*)d_in[8];
  const float* sp_emb = (const float*)d_in[9];
  const float* enc_W  = (const float*)d_in[10];
  const float* enc_b  = (const float*)d_in[11];
  const float* Wih    = (const float*)d_in[12];
  const float* Whh    = (const float*)d_in[13];
  const float* bih    = (const float*)d_in[14];
  const float* bhh    = (const float*)d_in[15];
  const float* postW  = (const float*)d_in[16];
  const float* postb  = (const float*)d_in[17];
  float* out = (float*)d_out;

  // workspace layout (fp32)
  float* ws     = (float*)d_ws;
  float* enc_in = ws;                                   // L*516
  float* h      = enc_in + (size_t)L_SEQ * CIN;         // L*1024
  float* gi     = h      + (size_t)L_SEQ * HDIM;        // L*3072
  float* g_hbuf = gi     + (size_t)L_SEQ * G3;          // 2*1024
  float* g_f0   = g_hbuf + 2 * HDIM;                    // 1 (+pad)
  unsigned* bar = (unsigned*)(g_f0 + 16);

  init_state<<<1, HDIM, 0, stream>>>(g_hbuf, g_f0, bar);

  build_enc_in<<<L_SEQ, 256, 0, stream>>>(vow, con, sa, ea, sap, eap, spkid,
                                          ph_emb, sp_emb, enc_in);

  // h = enc_in @ enc_W.T + enc_b          (M=8192, N=1024, K=516)
  wmma_gemm_nt<<<dim3(L_SEQ / 64, HDIM / 256), 256, 0, stream>>>(
      enc_in, CIN, enc_W, CIN, enc_b, h, HDIM, CIN);

  // gi = h @ Wih[:, :H].T                 (M=8192, N=3072, K=1024)
  wmma_gemm_nt<<<dim3(L_SEQ / 64, G3 / 256), 256, 0, stream>>>(
      h, HDIM, Wih, LDW_IH, nullptr, gi, G3, HDIM);

  const size_t smem = (size_t)(3 * UNITS_PER_WG * USTRIDE + HDIM) * sizeof(float);
  gru_scan<<<NWG, TPW, smem, stream>>>(gi, Whh, Wih, bih, bhh, postW, postb,
                                       g_hbuf, g_f0, bar, out);
}